// AnchorLoss_52716428591387
// MI455X (gfx1250) — compile-verified
//
#include <hip/hip_runtime.h>

typedef __attribute__((ext_vector_type(2))) float v2f;
typedef __attribute__((ext_vector_type(8))) float v8f;

#define BATCH 8
#define NPTS  2048
#define TILES (NPTS / 16)          // 128 tiles per dimension
#define INV_TEMP 0.1f              // 1 / TEMPERATURE

// ---------------------------------------------------------------------------
// Prep: P = embedding + abs_coords (D=2), Sq = |P|^2, into workspace arrays.
// ---------------------------------------------------------------------------
__global__ void anchor_prep_kernel(const float* __restrict__ emb,
                                   const float* __restrict__ coords,
                                   float* __restrict__ Px,
                                   float* __restrict__ Py,
                                   float* __restrict__ Sq,
                                   int total) {
    int i = blockIdx.x * blockDim.x + threadIdx.x;
    if (i < total) {
        float x = emb[2 * i]     + coords[2 * i];
        float y = emb[2 * i + 1] + coords[2 * i + 1];
        Px[i] = x;
        Py[i] = y;
        Sq[i] = x * x + y * y;
    }
}

// ---------------------------------------------------------------------------
// Main: one wave per 16x16 tile of the NxN pairwise-distance matrix.
// dist^2 computed as a single V_WMMA_F32_16X16X4_F32:
//   A row m = [-2*Px[m], -2*Py[m], Sq[m], 1]   (16x4)
//   B col n = [ Px[n],    Py[n],   1, Sq[n]]   (4x16)
//   D = A*B = Sq[m] + Sq[n] - 2*(p_m . p_n) = |p_m - p_n|^2
// ---------------------------------------------------------------------------
__global__ void __launch_bounds__(256)
anchor_loss_kernel(const float* __restrict__ Px,
                   const float* __restrict__ Py,
                   const float* __restrict__ Sq,
                   const int*   __restrict__ mask,
                   float*       __restrict__ out) {
    __shared__ float lds_sum;
    if (threadIdx.x == 0) lds_sum = 0.0f;
    __syncthreads();

    const int lane        = threadIdx.x & 31;
    const int waveInBlock = threadIdx.x >> 5;
    const int wavesPerBlk = blockDim.x >> 5;

    // Global tile id -> (batch, tileRow I, tileCol J). Grid sized exactly.
    int tile = blockIdx.x * wavesPerBlk + waveInBlock;
    int b    = tile / (TILES * TILES);
    int rem  = tile - b * (TILES * TILES);
    int I    = rem / TILES;
    int J    = rem - I * TILES;

    const int  lo      = lane & 15;
    const int  hi      = lane >> 4;          // 0: K=0,1 half; 1: K=2,3 half
    const bool lowHalf = (hi == 0);
    const int  base    = b * NPTS;
    const int  row     = base + I * 16 + lo; // M index for this lane (both halves)
    const int  col     = base + J * 16 + lo; // N index for this lane (both halves)

    // Small, cache-resident reads (Px/Py/Sq total 192 KB for all batches).
    float pxr = Px[row], pyr = Py[row], sqr = Sq[row];
    float pxc = Px[col], pyc = Py[col], sqc = Sq[col];

    // ISA 16x4 f32 A layout: lanes 0-15 hold {K0,K1}, lanes 16-31 hold {K2,K3}.
    v2f A, Bv;
    A.x  = lowHalf ? (-2.0f * pxr) : sqr;    // K0 = -2px   | K2 = |p_m|^2
    A.y  = lowHalf ? (-2.0f * pyr) : 1.0f;   // K1 = -2py   | K3 = 1
    Bv.x = lowHalf ? pxc           : 1.0f;   // K0 =  px    | K2 = 1
    Bv.y = lowHalf ? pyc           : sqc;    // K1 =  py    | K3 = |p_n|^2

    v8f C = {};
    // 8 args: (neg_a, A, neg_b, B, c_mod, C, reuse_a, reuse_b)
    C = __builtin_amdgcn_wmma_f32_16x16x4_f32(
        /*neg_a=*/false, A, /*neg_b=*/false, Bv,
        /*c_mod=*/(short)0, C, /*reuse_a=*/false, /*reuse_b=*/false);

    // C/D layout: VGPR r holds (M = r + 8*hi, N = lo). Mask tile loads:
    // each global_load_b32 instruction covers two contiguous 64B rows -> coalesced.
    const int* mrow = mask + (long long)(base + I * 16 + 8 * hi) * NPTS + (J * 16 + lo);
    float acc = 0.0f;
#pragma unroll
    for (int r = 0; r < 8; ++r) {
        int   mv  = mrow[(long long)r * NPTS];
        float d2  = C[r];
        float val = 1.0f - __expf(-INV_TEMP * d2);
        acc += (mv == 1) ? val : 0.0f;
    }

    // Wave32 butterfly reduction.
#pragma unroll
    for (int off = 16; off > 0; off >>= 1)
        acc += __shfl_xor(acc, off, 32);

    if (lane == 0) atomicAdd(&lds_sum, acc);   // ds_add_f32, 8 per block
    __syncthreads();
    if (threadIdx.x == 0) atomicAdd(out, lds_sum); // 16384 global atomics total
}

// ---------------------------------------------------------------------------
// Launcher
// ---------------------------------------------------------------------------
extern "C" void kernel_launch(void* const* d_in, const int* in_sizes, int n_in,
                              void* d_out, int out_size, void* d_ws, size_t ws_size,
                              hipStream_t stream) {
    const float* emb    = (const float*)d_in[0];  // [B, N, 2]
    const float* coords = (const float*)d_in[1];  // [B, N, 2]
    const int*   mask   = (const int*)  d_in[2];  // [B, N, N]
    float*       out    = (float*)d_out;          // scalar

    const int total = BATCH * NPTS;               // 16384 points
    float* Px = (float*)d_ws;
    float* Py = Px + total;
    float* Sq = Py + total;

    hipMemsetAsync(d_out, 0, sizeof(float) * out_size, stream);

    anchor_prep_kernel<<<(total + 255) / 256, 256, 0, stream>>>(
        emb, coords, Px, Py, Sq, total);

    const int tilesTotal = BATCH * TILES * TILES; // 131072 tiles, 1 wave each
    const int wavesPerBlock = 8;                  // 256 threads
    anchor_loss_kernel<<<tilesTotal / wavesPerBlock, 256, 0, stream>>>(
        Px, Py, Sq, mask, out);
}